// QTAttB_21620865368154
// MI455X (gfx1250) — compile-verified
//
#include <hip/hip_runtime.h>

typedef float v2f __attribute__((ext_vector_type(2)));
typedef float v8f __attribute__((ext_vector_type(8)));

// V_WMMA_F32_16X16X4_F32: D(16x16 f32) = A(16x4 f32) * B(4x16 f32) + C
// 8-arg pattern per probe: (neg_a, A, neg_b, B, c_mod, C, reuse_a, reuse_b)
__device__ __forceinline__ v8f wmma_f32_k4(v2f a, v2f b, v8f c) {
    return __builtin_amdgcn_wmma_f32_16x16x4_f32(false, a, false, b, (short)0, c, false, false);
}

#define NHEADS 8
#define DHEAD  16

// ---------------------------------------------------------------------------
// Coarse level: full attention over 32x32 tokens. One wave per (b, h, 32-query
// block). Scores via f32 WMMA -> LDS; lane-per-row softmax + top-16; AV via WMMA.
// ---------------------------------------------------------------------------
__global__ __launch_bounds__(32) void qta_coarse(const float* __restrict__ q,
                                                 const float* __restrict__ k,
                                                 const float* __restrict__ v,
                                                 float* __restrict__ msg0,
                                                 int* __restrict__ top0) {
    const int HW = 1024;   // 32*32 tokens
    const int TOPK = 16;

    const int bid  = blockIdx.x;
    const int qblk = bid & 31;         // 32 query blocks of 32 queries
    const int h    = (bid >> 5) & 7;
    const int b    = bid >> 8;

    const int lane = threadIdx.x;
    const int half = lane >> 4;        // 0: K-chunk {0,1}, 1: K-chunk {2,3}
    const int l16  = lane & 15;

    const float* qb = q + ((size_t)(b * 128) + h * DHEAD) * HW;
    const float* kb = k + ((size_t)(b * 128) + h * DHEAD) * HW;
    const float* vb = v + ((size_t)(b * 128) + h * DHEAD) * HW;
    const int q0 = qblk * 32;

    __shared__ float sc[32][1025];     // pad column 1024 stores 1/rowsum

    // ---- load Q as WMMA A-operand: 2 query tiles x 4 K-chunks of 4 dims ----
    v2f A[2][4];
#pragma unroll
    for (int t = 0; t < 2; ++t) {
#pragma unroll
        for (int c = 0; c < 4; ++c) {
            const int m  = q0 + t * 16 + l16;
            const int d0 = c * 4 + half * 2;
            v2f a;
            a.x = qb[(size_t)d0 * HW + m];
            a.y = qb[(size_t)(d0 + 1) * HW + m];
            A[t][c] = a;
        }
    }

    // ---- scores: S = (Q K^T) * d^-0.5  -> LDS ----
    for (int kt = 0; kt < 64; ++kt) {
        const int s0 = kt * 16;
        v8f acc0 = {0.f, 0.f, 0.f, 0.f, 0.f, 0.f, 0.f, 0.f};
        v8f acc1 = {0.f, 0.f, 0.f, 0.f, 0.f, 0.f, 0.f, 0.f};
#pragma unroll
        for (int c = 0; c < 4; ++c) {
            const int d0 = c * 4 + half * 2;
            v2f Bv;                                   // B[kd][n] = K[token n][dim]
            Bv.x = kb[(size_t)d0 * HW + s0 + l16];
            Bv.y = kb[(size_t)(d0 + 1) * HW + s0 + l16];
            acc0 = wmma_f32_k4(A[0][c], Bv, acc0);
            acc1 = wmma_f32_k4(A[1][c], Bv, acc1);
        }
        // D layout: VGPR r -> row (half*8 + r), col l16
#pragma unroll
        for (int r = 0; r < 8; ++r) {
            sc[half * 8 + r][s0 + l16]      = acc0[r] * 0.25f;
            sc[16 + half * 8 + r][s0 + l16] = acc1[r] * 0.25f;
        }
    }
    __syncthreads();

    // ---- per-row (lane-per-query) top-16 on raw scores ----
    float tv[16];
    int   ti[16];
#pragma unroll
    for (int j = 0; j < 16; ++j) { tv[j] = -3.0e38f; ti[j] = 0; }
    float* rowp = sc[lane];
    for (int i = 0; i < 1024; ++i) {
        const float x = rowp[i];
        if (x > tv[15]) {              // strict > keeps earlier-index-first ties
            float cv = x; int ci = i;
#pragma unroll
            for (int j = 0; j < 16; ++j) {
                const bool sw = cv > tv[j];
                const float nv = sw ? cv : tv[j];
                const int   ni = sw ? ci : ti[j];
                cv = sw ? tv[j] : cv;
                ci = sw ? ti[j] : ci;
                tv[j] = nv; ti[j] = ni;
            }
        }
    }
    // ---- softmax: exp in place, keep 1/sum in pad column ----
    const float mx = tv[0];
    float sum = 0.f;
    for (int i = 0; i < 1024; ++i) {
        const float e = __expf(rowp[i] - mx);
        rowp[i] = e;
        sum += e;
    }
    rowp[1024] = 1.0f / sum;

    int* tout = top0 + (((size_t)(b * NHEADS) + h) * 1024 + (q0 + lane)) * TOPK;
#pragma unroll
    for (int j = 0; j < 16; ++j) tout[j] = ti[j];
    __syncthreads();

    // ---- msg = softmax(S) @ V via WMMA ----
    v8f o0 = {0.f, 0.f, 0.f, 0.f, 0.f, 0.f, 0.f, 0.f};
    v8f o1 = {0.f, 0.f, 0.f, 0.f, 0.f, 0.f, 0.f, 0.f};
    for (int c = 0; c < 256; ++c) {    // 4 keys per chunk
        const int kb4 = c * 4;
        v2f Bv;                        // B[kd][n] = V[token kb4+kd][dim n]
        Bv.x = vb[(size_t)l16 * HW + kb4 + half * 2];
        Bv.y = vb[(size_t)l16 * HW + kb4 + half * 2 + 1];
        v2f P0, P1;                    // A[m][kd] = P[row m][key kb4+kd]
        P0.x = sc[l16][kb4 + half * 2];
        P0.y = sc[l16][kb4 + half * 2 + 1];
        P1.x = sc[16 + l16][kb4 + half * 2];
        P1.y = sc[16 + l16][kb4 + half * 2 + 1];
        o0 = wmma_f32_k4(P0, Bv, o0);
        o1 = wmma_f32_k4(P1, Bv, o1);
    }
    // store msg0 [b][tok][h][d], normalizing by 1/rowsum
#pragma unroll
    for (int r = 0; r < 8; ++r) {
        const int m0 = half * 8 + r;
        const int m1 = 16 + half * 8 + r;
        msg0[(((size_t)b * 1024 + q0 + m0) * NHEADS + h) * DHEAD + l16] = o0[r] * sc[m0][1024];
        msg0[(((size_t)b * 1024 + q0 + m1) * NHEADS + h) * DHEAD + l16] = o1[r] * sc[m1][1024];
    }
}

// ---------------------------------------------------------------------------
// Fine level: one thread per (b, group, head, quadrant). Gather 4*KPREV keys
// (children of previous winners), d=16 dots, softmax, top-k, weighted V sum.
// ---------------------------------------------------------------------------
template <int KPREV, int TOPK, bool WRITE_TOP>
__global__ __launch_bounds__(32) void qta_fine(const float* __restrict__ q,
                                               const float* __restrict__ k,
                                               const float* __restrict__ v,
                                               const int* __restrict__ ptop,
                                               float* __restrict__ msg,
                                               int* __restrict__ otop,
                                               int W) {
    const int K  = KPREV * 4;
    const int Wg = W >> 1;           // group-grid width == previous level width
    const int Lg = Wg * Wg;
    const int HW = W * W;

    const int tid = blockIdx.x * 32 + threadIdx.x;
    const int t = tid & 3;
    const int h = (tid >> 2) & 7;
    const int g = (tid >> 5) % Lg;
    const int b = tid / (Lg * 32);

    const int gy = g / Wg, gx = g % Wg;
    const int qtok = (2 * gy + (t >> 1)) * W + 2 * gx + (t & 1);

    const float* qb = q + ((size_t)(b * 128) + h * DHEAD) * HW;
    const float* kb = k + ((size_t)(b * 128) + h * DHEAD) * HW;
    const float* vb = v + ((size_t)(b * 128) + h * DHEAD) * HW;

    float qv[DHEAD];
#pragma unroll
    for (int dd = 0; dd < DHEAD; ++dd) qv[dd] = qb[(size_t)dd * HW + qtok];

    // previous winners -> child base coordinates at this level
    int py[KPREV], px[KPREV];
    const int* pt = ptop + (((size_t)(b * NHEADS) + h) * (size_t)Lg + g) * KPREV;
#pragma unroll
    for (int j = 0; j < KPREV; ++j) {
        const int pi = pt[j];
        py[j] = (pi / Wg) * 2;
        px[j] = (pi % Wg) * 2;
    }

    // ---- scores ----
    float s[K];
#pragma unroll
    for (int j = 0; j < KPREV; ++j) {
#pragma unroll
        for (int c = 0; c < 4; ++c) {
            const int si = (py[j] + (c >> 1)) * W + px[j] + (c & 1);
            float acc = 0.f;
#pragma unroll
            for (int dd = 0; dd < DHEAD; ++dd) acc += qv[dd] * kb[(size_t)dd * HW + si];
            s[j * 4 + c] = acc * 0.25f;
        }
    }

    // ---- top-k (store absolute source index directly) ----
    float tv[TOPK];
    int   ti[TOPK];
#pragma unroll
    for (int j = 0; j < TOPK; ++j) { tv[j] = -3.0e38f; ti[j] = 0; }
#pragma unroll
    for (int j = 0; j < KPREV; ++j) {
#pragma unroll
        for (int c = 0; c < 4; ++c) {
            const float x = s[j * 4 + c];
            if (x > tv[TOPK - 1]) {
                float cv = x;
                int   ci = (py[j] + (c >> 1)) * W + px[j] + (c & 1);
#pragma unroll
                for (int jj = 0; jj < TOPK; ++jj) {
                    const bool sw = cv > tv[jj];
                    const float nv = sw ? cv : tv[jj];
                    const int   ni = sw ? ci : ti[jj];
                    cv = sw ? tv[jj] : cv;
                    ci = sw ? ti[jj] : ci;
                    tv[jj] = nv; ti[jj] = ni;
                }
            }
        }
    }

    // ---- softmax in place ----
    const float mx = tv[0];
    float sum = 0.f;
#pragma unroll
    for (int kk = 0; kk < K; ++kk) { s[kk] = __expf(s[kk] - mx); sum += s[kk]; }
    const float inv = 1.0f / sum;

    // ---- msg = A @ gathered V ----
    float macc[DHEAD];
#pragma unroll
    for (int dd = 0; dd < DHEAD; ++dd) macc[dd] = 0.f;
#pragma unroll
    for (int j = 0; j < KPREV; ++j) {
#pragma unroll
        for (int c = 0; c < 4; ++c) {
            const int si = (py[j] + (c >> 1)) * W + px[j] + (c & 1);
            const float a = s[j * 4 + c];
#pragma unroll
            for (int dd = 0; dd < DHEAD; ++dd) macc[dd] += a * vb[(size_t)dd * HW + si];
        }
    }
    float* mo = msg + ((((size_t)b * Lg + g) * 4 + t) * NHEADS + h) * DHEAD;
#pragma unroll
    for (int dd = 0; dd < DHEAD; ++dd) mo[dd] = macc[dd] * inv;

    if (WRITE_TOP) {
        int* to = otop + (((size_t)(b * NHEADS) + h) * (size_t)HW + qtok) * TOPK;
#pragma unroll
        for (int j = 0; j < TOPK; ++j) to[j] = ti[j];
    }
}

// ---------------------------------------------------------------------------
// Combine: out[b, tok128, h, d] = w0*msg0(up4) + w1*msg1(up2) + w2*msg2
// ---------------------------------------------------------------------------
__global__ void qta_combine(const float* __restrict__ wgt,
                            const float* __restrict__ m0,
                            const float* __restrict__ m1,
                            const float* __restrict__ m2,
                            float* __restrict__ out, size_t total) {
    const size_t i = (size_t)blockIdx.x * blockDim.x + threadIdx.x;
    if (i >= total) return;
    const int dd  = (int)(i & 15);
    const int h   = (int)((i >> 4) & 7);
    const int tok = (int)((i >> 7) & 16383);
    const int b   = (int)(i >> 21);
    const int Y = tok >> 7, X = tok & 127;
    const int y64 = Y >> 1, x64 = X >> 1;
    const int qd2 = (Y & 1) * 2 + (X & 1);
    const int qd1 = ((Y >> 1) & 1) * 2 + ((X >> 1) & 1);
    const int tok32 = (Y >> 2) * 32 + (X >> 2);
    const int g128  = y64 * 64 + x64;

    const float w0r = wgt[0], w1r = wgt[1], w2r = wgt[2];
    const float wm = fmaxf(w0r, fmaxf(w1r, w2r));
    const float e0 = __expf(w0r - wm), e1 = __expf(w1r - wm), e2 = __expf(w2r - wm);
    const float einv = 1.0f / (e0 + e1 + e2);

    const float a0 = m0[(((size_t)b * 1024 + tok32) * NHEADS + h) * DHEAD + dd];
    const float a1 = m1[((((size_t)b * 1024 + tok32) * 4 + qd1) * NHEADS + h) * DHEAD + dd];
    const float a2 = m2[((((size_t)b * 4096 + g128) * 4 + qd2) * NHEADS + h) * DHEAD + dd];
    out[i] = (a0 * e0 + a1 * e1 + a2 * e2) * einv;
}

extern "C" void kernel_launch(void* const* d_in, const int* in_sizes, int n_in,
                              void* d_out, int out_size, void* d_ws, size_t ws_size,
                              hipStream_t stream) {
    const float* q0 = (const float*)d_in[0];
    const float* q1 = (const float*)d_in[1];
    const float* q2 = (const float*)d_in[2];
    const float* k0 = (const float*)d_in[3];
    const float* k1 = (const float*)d_in[4];
    const float* k2 = (const float*)d_in[5];
    const float* v0 = (const float*)d_in[6];
    const float* v1 = (const float*)d_in[7];
    const float* v2 = (const float*)d_in[8];
    const float* wgt = (const float*)d_in[9];
    (void)in_sizes; (void)n_in; (void)out_size; (void)ws_size;

    char* ws = (char*)d_ws;
    size_t off = 0;
    float* msg0 = (float*)(ws + off); off += (size_t)4 * 1024 * 8 * 16 * 4;        // 2 MB
    int*   top0 = (int*)(ws + off);   off += (size_t)4 * 8 * 1024 * 16 * 4;        // 2 MB
    float* msg1 = (float*)(ws + off); off += (size_t)4 * 1024 * 4 * 8 * 16 * 4;    // 8 MB
    int*   top1 = (int*)(ws + off);   off += (size_t)4 * 8 * 4096 * 8 * 4;         // 4 MB
    float* msg2 = (float*)(ws + off); off += (size_t)4 * 4096 * 4 * 8 * 16 * 4;    // 32 MB

    // Level 0 (coarse, 32x32): 4 b * 8 h * 32 query-blocks = 1024 waves
    qta_coarse<<<1024, 32, 0, stream>>>(q2, k2, v2, msg0, top0);

    // Level 1 (64x64): threads = 4*1024*8*4 = 131072
    qta_fine<16, 8, true><<<131072 / 32, 32, 0, stream>>>(q1, k1, v1, top0, msg1, top1, 64);

    // Level 2 (128x128): threads = 4*4096*8*4 = 524288 (top-k of last level unused)
    qta_fine<8, 8, false><<<524288 / 32, 32, 0, stream>>>(q0, k0, v0, top1, msg2, nullptr, 128);

    // Combine: 4*16384*8*16 = 8388608 outputs
    const size_t total = (size_t)4 * 16384 * 8 * 16;
    qta_combine<<<(unsigned)((total + 255) / 256), 256, 0, stream>>>(wgt, msg0, msg1, msg2,
                                                                     (float*)d_out, total);
}